// BN1dFitlered_54331336294643
// MI455X (gfx1250) — compile-verified
//
#include <hip/hip_runtime.h>
#include <math.h>

// ---------------------------------------------------------------------------
// BN1d with outlier-filtered statistics, MI455X (gfx1250, wave32).
// Pass1 (fwd): Sx, Sxx via in-loop V_WMMA_F32_16X16X4_F32 ones-matrix folds.
// Pass2 (bwd, rides L2): masked Sx, Sxx, cnt  (mask = lo < x < hi).
// Pass3 (fwd, rides L2): out = x*scale + shift, NT loads + NT stores.
// ---------------------------------------------------------------------------

typedef float v2f __attribute__((ext_vector_type(2)));
typedef float v4f __attribute__((ext_vector_type(4)));
typedef float v8f __attribute__((ext_vector_type(8)));

#define THRES_D     4.0
#define EPS_D       1e-10
#define RED_BLOCKS  1024
#define BLOCK_T     256
#define OUT_BLOCKS  8192

// D = ones(16x4) * B(4x16) + C : every D[m][n] = colsum_n(B) + C[m][n].
// Feeding per-lane data as B folds 64 fp32 (2 VGPRs x 32 lanes) into 16
// replicated column sums per instruction, accumulated in C.
__device__ __forceinline__ v8f wmma_colsum_acc(v2f b, v8f c) {
  v2f ones = {1.0f, 1.0f};
  return __builtin_amdgcn_wmma_f32_16x16x4_f32(false, ones, false, b,
                                               (short)0, c, false, false);
}

// D rows are all identical (all-ones A, zero-init C), so lanes 0..15 of d[0]
// hold colsum[0..15] and lanes 16..31 duplicate them. Summing within each
// 16-lane half yields the wave total.
__device__ __forceinline__ float wave_total_from_d(v8f d) {
  float s = d[0];
  s += __shfl_xor(s, 1, 32);
  s += __shfl_xor(s, 2, 32);
  s += __shfl_xor(s, 4, 32);
  s += __shfl_xor(s, 8, 32);
  return s;
}

__device__ __forceinline__ float wave_total_f(float s) {
  s += __shfl_xor(s, 1, 32);
  s += __shfl_xor(s, 2, 32);
  s += __shfl_xor(s, 4, 32);
  s += __shfl_xor(s, 8, 32);
  s += __shfl_xor(s, 16, 32);
  return s;
}

// ---------------------------------------------------------------------------
// Pass 1: forward streaming reduction of Sx, Sxx.
// 256 floats / wave / chunk = two b128 loads in flight per wave.
// ---------------------------------------------------------------------------
__global__ void __launch_bounds__(BLOCK_T)
pass1_reduce(const float* __restrict__ x, double* __restrict__ part, long n) {
  const int  lane = threadIdx.x & 31;
  const int  wib  = __builtin_amdgcn_readfirstlane(threadIdx.x >> 5); // scalar
  const int  wpb  = BLOCK_T >> 5;
  const long wid    = (long)blockIdx.x * wpb + wib;                   // scalar
  const long nwaves = (long)gridDim.x * wpb;
  const long nchunk = n >> 8;                 // 256 floats / wave / chunk
  const v4f* __restrict__ x4 = (const v4f*)x;

  v8f ax0 = 0.0f, ax1 = 0.0f, aq0 = 0.0f, aq1 = 0.0f;
  for (long c = wid; c < nchunk; c += nwaves) {       // scalar loop: EXEC full
    const long base = c << 6;                         // v4f units
    v4f va = x4[base + lane];                         // global_load_b128
    v4f vb = x4[base + 32 + lane];                    // global_load_b128
    v2f b0 = {va.x, va.y};
    v2f b1 = {va.z, va.w};
    v2f b2 = {vb.x, vb.y};
    v2f b3 = {vb.z, vb.w};
    v2f q0 = {va.x * va.x, va.y * va.y};
    v2f q1 = {va.z * va.z, va.w * va.w};
    v2f q2 = {vb.x * vb.x, vb.y * vb.y};
    v2f q3 = {vb.z * vb.z, vb.w * vb.w};
    ax0 = wmma_colsum_acc(b0, ax0);                   // 4 chains, depth 2
    ax1 = wmma_colsum_acc(b1, ax1);
    aq0 = wmma_colsum_acc(q0, aq0);
    aq1 = wmma_colsum_acc(q1, aq1);
    ax0 = wmma_colsum_acc(b2, ax0);
    ax1 = wmma_colsum_acc(b3, ax1);
    aq0 = wmma_colsum_acc(q2, aq0);
    aq1 = wmma_colsum_acc(q3, aq1);
  }
  float tsx = 0.0f, tsq = 0.0f;                        // scalar tail (n % 256)
  for (long i = (nchunk << 8) + wid * 32 + lane; i < n; i += nwaves * 32) {
    float v = x[i];
    tsx += v; tsq += v * v;
  }
  float wsx = wave_total_from_d(ax0) + wave_total_from_d(ax1) + wave_total_f(tsx);
  float wsq = wave_total_from_d(aq0) + wave_total_from_d(aq1) + wave_total_f(tsq);

  __shared__ float sX[8], sQ[8];
  if (lane == 0) { sX[wib] = wsx; sQ[wib] = wsq; }
  __syncthreads();
  if (threadIdx.x == 0) {
    double bx = 0.0, bq = 0.0;
    for (int i = 0; i < wpb; ++i) { bx += (double)sX[i]; bq += (double)sQ[i]; }
    part[2 * (long)blockIdx.x + 0] = bx;               // deterministic slot
    part[2 * (long)blockIdx.x + 1] = bq;
  }
}

// Final pass-1 reduction + mask thresholds: |(x-m1)*rsqrt(v1+eps)| < 4
// <=>  m1 - 4*sqrt(v1+eps) < x < m1 + 4*sqrt(v1+eps).
__global__ void __launch_bounds__(BLOCK_T)
pass1_finalize(const double* __restrict__ part, float* __restrict__ scal,
               long n, int npart) {
  __shared__ double s0[BLOCK_T], s1[BLOCK_T];
  double a = 0.0, b = 0.0;
  for (int i = threadIdx.x; i < npart; i += BLOCK_T) {
    a += part[2 * i]; b += part[2 * i + 1];
  }
  s0[threadIdx.x] = a; s1[threadIdx.x] = b;
  __syncthreads();
  for (int off = BLOCK_T >> 1; off > 0; off >>= 1) {
    if (threadIdx.x < off) {
      s0[threadIdx.x] += s0[threadIdx.x + off];
      s1[threadIdx.x] += s1[threadIdx.x + off];
    }
    __syncthreads();
  }
  if (threadIdx.x == 0) {
    double sx = s0[0], sq = s1[0];
    double mean1 = sx / (double)n;
    double var1  = (sq - (double)n * mean1 * mean1) / (double)(n - 1);
    double t     = THRES_D * sqrt(var1 + EPS_D);
    scal[0] = (float)(mean1 - t);
    scal[1] = (float)(mean1 + t);
  }
}

// ---------------------------------------------------------------------------
// Pass 2: BACKWARD masked reduction (hits the ~192MB tail pass 1 left in L2).
// ---------------------------------------------------------------------------
__global__ void __launch_bounds__(BLOCK_T)
pass2_reduce(const float* __restrict__ x, const float* __restrict__ scal,
             double* __restrict__ part, long n) {
  const float lo = scal[0], hi = scal[1];
  const int  lane = threadIdx.x & 31;
  const int  wib  = __builtin_amdgcn_readfirstlane(threadIdx.x >> 5); // scalar
  const int  wpb  = BLOCK_T >> 5;
  const long wid    = (long)blockIdx.x * wpb + wib;
  const long nwaves = (long)gridDim.x * wpb;
  const long nchunk = n >> 8;
  const v4f* __restrict__ x4 = (const v4f*)x;

  v8f ax0 = 0.0f, ax1 = 0.0f, aq0 = 0.0f, aq1 = 0.0f;
  float cnt = 0.0f;
  for (long c = nchunk - 1 - wid; c >= 0; c -= nwaves) {   // reverse walk
    const long base = c << 6;
    v4f va = x4[base + lane];
    v4f vb = x4[base + 32 + lane];
    float ma0 = (va.x > lo && va.x < hi) ? 1.0f : 0.0f;
    float ma1 = (va.y > lo && va.y < hi) ? 1.0f : 0.0f;
    float ma2 = (va.z > lo && va.z < hi) ? 1.0f : 0.0f;
    float ma3 = (va.w > lo && va.w < hi) ? 1.0f : 0.0f;
    float mb0 = (vb.x > lo && vb.x < hi) ? 1.0f : 0.0f;
    float mb1 = (vb.y > lo && vb.y < hi) ? 1.0f : 0.0f;
    float mb2 = (vb.z > lo && vb.z < hi) ? 1.0f : 0.0f;
    float mb3 = (vb.w > lo && vb.w < hi) ? 1.0f : 0.0f;
    v2f b0 = {va.x * ma0, va.y * ma1};
    v2f b1 = {va.z * ma2, va.w * ma3};
    v2f b2 = {vb.x * mb0, vb.y * mb1};
    v2f b3 = {vb.z * mb2, vb.w * mb3};
    v2f q0 = {(va.x * va.x) * ma0, (va.y * va.y) * ma1};
    v2f q1 = {(va.z * va.z) * ma2, (va.w * va.w) * ma3};
    v2f q2 = {(vb.x * vb.x) * mb0, (vb.y * vb.y) * mb1};
    v2f q3 = {(vb.z * vb.z) * mb2, (vb.w * vb.w) * mb3};
    cnt += (ma0 + ma1 + ma2 + ma3) + (mb0 + mb1 + mb2 + mb3);
    ax0 = wmma_colsum_acc(b0, ax0);
    ax1 = wmma_colsum_acc(b1, ax1);
    aq0 = wmma_colsum_acc(q0, aq0);
    aq1 = wmma_colsum_acc(q1, aq1);
    ax0 = wmma_colsum_acc(b2, ax0);
    ax1 = wmma_colsum_acc(b3, ax1);
    aq0 = wmma_colsum_acc(q2, aq0);
    aq1 = wmma_colsum_acc(q3, aq1);
  }
  float tsx = 0.0f, tsq = 0.0f;
  for (long i = (nchunk << 8) + wid * 32 + lane; i < n; i += nwaves * 32) {
    float v = x[i];
    if (v > lo && v < hi) { tsx += v; tsq += v * v; cnt += 1.0f; }
  }
  float wsx = wave_total_from_d(ax0) + wave_total_from_d(ax1) + wave_total_f(tsx);
  float wsq = wave_total_from_d(aq0) + wave_total_from_d(aq1) + wave_total_f(tsq);
  float wc  = wave_total_f(cnt);

  __shared__ float sX[8], sQ[8], sC[8];
  if (lane == 0) { sX[wib] = wsx; sQ[wib] = wsq; sC[wib] = wc; }
  __syncthreads();
  if (threadIdx.x == 0) {
    double bx = 0.0, bq = 0.0, bc = 0.0;
    for (int i = 0; i < wpb; ++i) {
      bx += (double)sX[i]; bq += (double)sQ[i]; bc += (double)sC[i];
    }
    part[3 * (long)blockIdx.x + 0] = bx;
    part[3 * (long)blockIdx.x + 1] = bq;
    part[3 * (long)blockIdx.x + 2] = bc;
  }
}

// mean2 = Smx/cnt ; var2 = (Smxx - cnt*mean2^2)/(cnt-1) ; fold gamma/beta in.
__global__ void __launch_bounds__(BLOCK_T)
pass2_finalize(const double* __restrict__ part, const float* __restrict__ gamma,
               const float* __restrict__ beta, float* __restrict__ scal, int npart) {
  __shared__ double s0[BLOCK_T], s1[BLOCK_T], s2[BLOCK_T];
  double a = 0.0, b = 0.0, c = 0.0;
  for (int i = threadIdx.x; i < npart; i += BLOCK_T) {
    a += part[3 * i]; b += part[3 * i + 1]; c += part[3 * i + 2];
  }
  s0[threadIdx.x] = a; s1[threadIdx.x] = b; s2[threadIdx.x] = c;
  __syncthreads();
  for (int off = BLOCK_T >> 1; off > 0; off >>= 1) {
    if (threadIdx.x < off) {
      s0[threadIdx.x] += s0[threadIdx.x + off];
      s1[threadIdx.x] += s1[threadIdx.x + off];
      s2[threadIdx.x] += s2[threadIdx.x + off];
    }
    __syncthreads();
  }
  if (threadIdx.x == 0) {
    double cnt   = s2[0];
    double mean2 = s0[0] / cnt;
    double var2  = (s1[0] - cnt * mean2 * mean2) / (cnt - 1.0);
    double inv   = 1.0 / sqrt(var2 + EPS_D);
    double scale = (double)gamma[0] * inv;
    scal[2] = (float)scale;
    scal[3] = (float)((double)beta[0] - scale * mean2);
  }
}

// ---------------------------------------------------------------------------
// Pass 3: forward fused scale+shift (hits the ~192MB head pass 2 left in L2).
// NT loads (last use of x) + NT stores (streaming output, keep L2 clean).
// ---------------------------------------------------------------------------
__global__ void __launch_bounds__(BLOCK_T)
finalize_out(const float* __restrict__ x, const float* __restrict__ scal,
             float* __restrict__ out, long n) {
  const float scale = scal[2], shift = scal[3];
  const long nv     = n >> 2;
  const long tid    = (long)blockIdx.x * BLOCK_T + threadIdx.x;
  const long stride = (long)gridDim.x * BLOCK_T;
  const v4f* __restrict__ x4 = (const v4f*)x;
  v4f* __restrict__ o4 = (v4f*)out;
  for (long i = tid; i < nv; i += stride) {
    v4f v = __builtin_nontemporal_load(&x4[i]);
    v4f r;
    r.x = fmaf(v.x, scale, shift);
    r.y = fmaf(v.y, scale, shift);
    r.z = fmaf(v.z, scale, shift);
    r.w = fmaf(v.w, scale, shift);
    __builtin_nontemporal_store(r, &o4[i]);
  }
  for (long i = (nv << 2) + tid; i < n; i += stride) {
    out[i] = fmaf(x[i], scale, shift);
  }
}

// ---------------------------------------------------------------------------
extern "C" void kernel_launch(void* const* d_in, const int* in_sizes, int n_in,
                              void* d_out, int out_size, void* d_ws, size_t ws_size,
                              hipStream_t stream) {
  const float* x     = (const float*)d_in[0];
  const float* gamma = (const float*)d_in[1];
  const float* beta  = (const float*)d_in[2];
  float* out = (float*)d_out;
  const long n = (long)in_sizes[0];

  // Workspace: deterministic per-block partials (no atomics) + scalars.
  double* part1 = (double*)d_ws;                    // [RED_BLOCKS*2]
  double* part2 = part1 + 2 * RED_BLOCKS;           // [RED_BLOCKS*3]
  float*  scal  = (float*)(part2 + 3 * RED_BLOCKS); // lo, hi, scale, shift

  pass1_reduce  <<<RED_BLOCKS, BLOCK_T, 0, stream>>>(x, part1, n);
  pass1_finalize<<<1,          BLOCK_T, 0, stream>>>(part1, scal, n, RED_BLOCKS);
  pass2_reduce  <<<RED_BLOCKS, BLOCK_T, 0, stream>>>(x, scal, part2, n);
  pass2_finalize<<<1,          BLOCK_T, 0, stream>>>(part2, gamma, beta, scal, RED_BLOCKS);
  finalize_out  <<<OUT_BLOCKS, BLOCK_T, 0, stream>>>(x, scal, out, n);
}